// Hand_Optimizee_Model_rasa_78323023609973
// MI455X (gfx1250) — compile-verified
//
#include <hip/hip_runtime.h>

// Problem constants
#define NB   64
#define DIN  1024
#define DOUT 256
#define LR2  1e-4f   // lr applied twice in the reference

#define KC   32      // K-chunk staged in LDS per iteration

typedef __attribute__((ext_vector_type(2))) float v2f;
typedef __attribute__((ext_vector_type(8))) float v8f;
typedef __attribute__((ext_vector_type(4))) int   v4i;

// Pointer-to-AS1-v4i and pointer-to-AS3-v4i (pointee carries the addr space)
typedef __attribute__((address_space(1))) v4i* g_v4i_ptr;
typedef __attribute__((address_space(3))) v4i* l_v4i_ptr;

__device__ __forceinline__ v8f wmma_f32(v2f a, v2f b, v8f c) {
  // V_WMMA_F32_16X16X4_F32 : D(16x16 f32) = A(16x4 f32) * B(4x16 f32) + C
  return __builtin_amdgcn_wmma_f32_16x16x4_f32(false, a, false, b, (short)0, c,
                                               false, false);
}

// 16-byte global -> LDS copy: async-to-LDS when available (gfx1250 ASYNCcnt
// path), else regular load + ds_store.
__device__ __forceinline__ void cp16(float* dst_lds, const float* src_glb) {
#if __has_builtin(__builtin_amdgcn_global_load_async_to_lds_b128)
  __builtin_amdgcn_global_load_async_to_lds_b128(
      (g_v4i_ptr)src_glb, (l_v4i_ptr)dst_lds, 0, 0);
#else
  *(float4*)dst_lds = *(const float4*)src_glb;
#endif
}

__device__ __forceinline__ void cp_wait_barrier() {
#if __has_builtin(__builtin_amdgcn_global_load_async_to_lds_b128)
#if __has_builtin(__builtin_amdgcn_s_wait_asynccnt)
  __builtin_amdgcn_s_wait_asynccnt(0);
#else
  asm volatile("s_wait_asynccnt 0x0" ::: "memory");
#endif
#endif
  __syncthreads();
}

// ---------------------------------------------------------------------------
// Kernel 1: per-row second moment; ll and ll^{-1/4}
// grid = NB*DIN blocks, 256 threads; block b handles row (n,i) = b
// ---------------------------------------------------------------------------
__global__ void k_rowstats(const float* __restrict__ grad,
                           const float* __restrict__ L0,
                           float* __restrict__ out_ll,
                           float* __restrict__ invL) {
  __shared__ float sbuf[256];
  const int b = blockIdx.x;
  const int tid = threadIdx.x;
  const float v = grad[(size_t)b * DOUT + tid];
  sbuf[tid] = v * v;
  __syncthreads();
  for (int off = 128; off > 0; off >>= 1) {
    if (tid < off) sbuf[tid] += sbuf[tid + off];
    __syncthreads();
  }
  if (tid == 0) {
    const float g2 = sbuf[0];
    const float l0 = L0[b];
    const float l = 0.5f * l0 + 0.5f * g2 * (1.0f / (float)DOUT);
    const float ll = fmaxf(l0, l);
    out_ll[b] = ll;
    invL[b] = __frsqrt_rn(__fsqrt_rn(ll));  // ll^{-1/4}
  }
}

// ---------------------------------------------------------------------------
// Kernel 2: per-column second moment; rr and rr^{-1/4}; also copies state
// ---------------------------------------------------------------------------
__global__ void k_colstats(const float* __restrict__ grad,
                           const float* __restrict__ R0,
                           float* __restrict__ out_rr,
                           float* __restrict__ invR,
                           const float* __restrict__ state,
                           float* __restrict__ out_state) {
  const int n = blockIdx.x;
  const int o = threadIdx.x;
  const float* g = grad + (size_t)n * DIN * DOUT + o;
  float s = 0.f;
  for (int i = 0; i < DIN; ++i) {
    const float v = g[(size_t)i * DOUT];
    s += v * v;
  }
  const float r0 = R0[n * DOUT + o];
  const float r = 0.5f * r0 + 0.5f * s * (1.0f / (float)DIN);
  const float rr = fmaxf(r0, r);
  out_rr[n * DOUT + o] = rr;
  invR[n * DOUT + o] = __frsqrt_rn(__fsqrt_rn(rr));
  if (n == 0 && o == 0) out_state[0] = state[0];
}

// ---------------------------------------------------------------------------
// Kernel 3: UU = grad * ll^{-1/4}[i] * rr^{-1/4}[o]
// ---------------------------------------------------------------------------
__global__ void k_scale(const float* __restrict__ grad,
                        const float* __restrict__ invL,
                        const float* __restrict__ invR,
                        float* __restrict__ UU) {
  const int b = blockIdx.x;          // n*DIN + i
  const int n = b >> 10;             // DIN = 1024
  const int o = threadIdx.x;
  const size_t idx = (size_t)b * DOUT + o;
  UU[idx] = grad[idx] * invL[b] * invR[n * DOUT + o];
}

// ---------------------------------------------------------------------------
// Kernel 4: C[n] = Amat[n]^T * Bmat[n]   (Amat,Bmat: [DIN x DOUT], C: 256x256)
// LDS-blocked: block = 256 thr (8 waves) computes a 128(P) x 64(O) C-block.
// K consumed in chunks of KC=32 staged into LDS (async-to-LDS when available).
// Wave w owns P-rows [w*16, w*16+16) and 4 O-tiles (A-fragment reused 4x).
// grid = NB * 8 blocks  (2 P-blocks x 4 O-blocks per n)
// ---------------------------------------------------------------------------
__global__ void k_gemm_tn(const float* __restrict__ Amat,
                          const float* __restrict__ Bmat,
                          float* __restrict__ C) {
  __shared__ __align__(16) float lds_A[KC][128];  // 16 KB
  __shared__ __align__(16) float lds_B[KC][64];   //  8 KB

  const int tid  = threadIdx.x;
  const int lane = tid & 31;
  const int w    = tid >> 5;        // wave id 0..7
  const int blk  = blockIdx.x;
  const int n    = blk >> 3;
  const int pb   = (blk & 7) >> 2;  // 0..1
  const int ob   = blk & 3;         // 0..3
  const int P0 = pb * 128, O0 = ob * 64;

  const int m  = lane & 15;
  const int kc = lane >> 4;

  const float* A = Amat + (size_t)n * DIN * DOUT;
  const float* B = Bmat + (size_t)n * DIN * DOUT;

  v8f acc[4];
  #pragma unroll
  for (int t = 0; t < 4; ++t) acc[t] = (v8f){0.f,0.f,0.f,0.f,0.f,0.f,0.f,0.f};

  for (int kk = 0; kk < DIN; kk += KC) {
    // ---- stage A panel: lds_A[k][p] = Amat[kk+k][P0+p], k<32, p<128 ----
    #pragma unroll
    for (int r = 0; r < 4; ++r) {
      const int s = tid + 256 * r;          // 1024 slots of 4 floats
      const int k = s >> 5, p4 = (s & 31) * 4;
      cp16(&lds_A[k][p4], A + (size_t)(kk + k) * DOUT + P0 + p4);
    }
    // ---- stage B panel: lds_B[k][o] = Bmat[kk+k][O0+o], k<32, o<64 ----
    #pragma unroll
    for (int r = 0; r < 2; ++r) {
      const int s = tid + 256 * r;          // 512 slots of 4 floats
      const int k = s >> 4, o4 = (s & 15) * 4;
      cp16(&lds_B[k][o4], B + (size_t)(kk + k) * DOUT + O0 + o4);
    }
    cp_wait_barrier();

    // ---- compute: 8 K-steps of 4, 4 WMMA per step per wave ----
    #pragma unroll
    for (int k4 = 0; k4 < KC; k4 += 4) {
      v2f a;
      a.x = lds_A[k4 + 2 * kc][w * 16 + m];
      a.y = lds_A[k4 + 2 * kc + 1][w * 16 + m];
      #pragma unroll
      for (int t = 0; t < 4; ++t) {
        v2f b;
        b.x = lds_B[k4 + 2 * kc][t * 16 + m];
        b.y = lds_B[k4 + 2 * kc + 1][t * 16 + m];
        acc[t] = wmma_f32(a, b, acc[t]);
      }
    }
    __syncthreads();
  }

  float* Cp = C + (size_t)n * DOUT * DOUT;
  #pragma unroll
  for (int t = 0; t < 4; ++t)
    #pragma unroll
    for (int r = 0; r < 8; ++r)
      Cp[(size_t)(P0 + w * 16 + r + 8 * kc) * DOUT + O0 + t * 16 + m] =
          acc[t][r];
}

// ---------------------------------------------------------------------------
// Kernel 5: D[n] = X[n] * (bscale*B[n])  (+ optional init P0 + c1*P1)
// X,D,P0,P1: [DIN x DOUT]; B: [DOUT x DOUT].
// LDS-blocked: block computes 128(I) x 64(O); K chunks of 32 staged in LDS.
// grid = NB * 32 blocks (8 I-blocks x 4 O-blocks per n)
// Used for: A = M - LR2*UU + M*(LR2*T)   and   Q = A * Rinv
// ---------------------------------------------------------------------------
__global__ void k_gemm_nn(const float* __restrict__ X,
                          const float* __restrict__ B,
                          const float* __restrict__ P0m,
                          const float* __restrict__ P1m,
                          float c1, float bscale, int use_init,
                          float* __restrict__ D) {
  __shared__ __align__(16) float lds_X[128][KC];  // 16 KB (row i, k fast)
  __shared__ __align__(16) float lds_B[KC][64];   //  8 KB

  const int tid  = threadIdx.x;
  const int lane = tid & 31;
  const int w    = tid >> 5;
  const int blk  = blockIdx.x;
  const int n    = blk >> 5;
  const int ib   = (blk & 31) >> 2;  // 0..7
  const int ob   = blk & 3;          // 0..3
  const int I0 = ib * 128, O0 = ob * 64;

  const int m  = lane & 15;
  const int kc = lane >> 4;

  const float* Xn = X + (size_t)n * DIN * DOUT;
  const float* Bn = B + (size_t)n * DOUT * DOUT;

  v8f acc[4];
  if (use_init) {
    const float* p0 = P0m + (size_t)n * DIN * DOUT;
    const float* p1 = P1m + (size_t)n * DIN * DOUT;
    #pragma unroll
    for (int t = 0; t < 4; ++t)
      #pragma unroll
      for (int r = 0; r < 8; ++r) {
        const size_t idx =
            (size_t)(I0 + w * 16 + r + 8 * kc) * DOUT + O0 + t * 16 + m;
        acc[t][r] = p0[idx] + c1 * p1[idx];
      }
  } else {
    #pragma unroll
    for (int t = 0; t < 4; ++t)
      acc[t] = (v8f){0.f,0.f,0.f,0.f,0.f,0.f,0.f,0.f};
  }

  for (int kk = 0; kk < DOUT; kk += KC) {
    // ---- stage X panel: lds_X[i][k] = X[I0+i][kk+k], i<128, k<32 ----
    #pragma unroll
    for (int r = 0; r < 4; ++r) {
      const int s = tid + 256 * r;          // 1024 slots of 4 floats
      const int i = s >> 3, k4 = (s & 7) * 4;
      cp16(&lds_X[i][k4], Xn + (size_t)(I0 + i) * DOUT + kk + k4);
    }
    // ---- stage B panel: lds_B[k][o] = B[kk+k][O0+o] ----
    #pragma unroll
    for (int r = 0; r < 2; ++r) {
      const int s = tid + 256 * r;
      const int k = s >> 4, o4 = (s & 15) * 4;
      cp16(&lds_B[k][o4], Bn + (size_t)(kk + k) * DOUT + O0 + o4);
    }
    cp_wait_barrier();

    #pragma unroll
    for (int k4 = 0; k4 < KC; k4 += 4) {
      const v2f a = *(const v2f*)&lds_X[w * 16 + m][k4 + 2 * kc];  // ds_b64
      #pragma unroll
      for (int t = 0; t < 4; ++t) {
        v2f b;
        b.x = bscale * lds_B[k4 + 2 * kc][t * 16 + m];
        b.y = bscale * lds_B[k4 + 2 * kc + 1][t * 16 + m];
        acc[t] = wmma_f32(a, b, acc[t]);
      }
    }
    __syncthreads();
  }

  float* Dp = D + (size_t)n * DIN * DOUT;
  #pragma unroll
  for (int t = 0; t < 4; ++t)
    #pragma unroll
    for (int r = 0; r < 8; ++r)
      Dp[(size_t)(I0 + w * 16 + r + 8 * kc) * DOUT + O0 + t * 16 + m] =
          acc[t][r];
}

// ---------------------------------------------------------------------------
// Kernel 6: per-batch Cholesky of G (upper triangle -> R, diag(R)>0) followed
// by upper-triangular inversion Rinv = R^{-1}. One block (256 thr) per n.
// ---------------------------------------------------------------------------
__global__ void k_cholinv(float* __restrict__ G, float* __restrict__ Rinv) {
  const int n = blockIdx.x;
  const int tid = threadIdx.x;
  float* g = G + (size_t)n * DOUT * DOUT;
  float* Xc = Rinv + (size_t)n * DOUT * DOUT;

  // In-place Cholesky: G = R^T R, R stored in upper triangle
  for (int j = 0; j < DOUT; ++j) {
    if (tid == 0) g[j * DOUT + j] = __fsqrt_rn(g[j * DOUT + j]);
    __syncthreads();
    const float rjj = g[j * DOUT + j];
    if (tid > j) g[j * DOUT + tid] /= rjj;
    __syncthreads();
    const float rjt = g[j * DOUT + tid];  // used only when tid > j
    for (int i = j + 1; i < DOUT; ++i) {
      const float rji = g[j * DOUT + i];
      if (tid >= i) g[i * DOUT + tid] -= rji * rjt;
    }
    __syncthreads();
  }

  // Column-parallel inversion of upper-triangular R: thread c owns column c
  const int cc = tid;
  for (int i = cc + 1; i < DOUT; ++i) Xc[i * DOUT + cc] = 0.f;  // lower zeros
  Xc[cc * DOUT + cc] = 1.0f / g[cc * DOUT + cc];
  for (int i = cc - 1; i >= 0; --i) {
    float s = 0.f;
    for (int k = i + 1; k <= cc; ++k) s += g[i * DOUT + k] * Xc[k * DOUT + cc];
    Xc[i * DOUT + cc] = -s / g[i * DOUT + i];
  }
}

// ---------------------------------------------------------------------------
// Host launcher
// ---------------------------------------------------------------------------
extern "C" void kernel_launch(void* const* d_in, const int* in_sizes, int n_in,
                              void* d_out, int out_size, void* d_ws,
                              size_t ws_size, hipStream_t stream) {
  (void)in_sizes; (void)n_in; (void)out_size; (void)ws_size;
  const float* grad  = (const float*)d_in[0];
  const float* M     = (const float*)d_in[1];
  const float* state = (const float*)d_in[2];
  const float* L0    = (const float*)d_in[3];
  const float* R0    = (const float*)d_in[4];

  float* out = (float*)d_out;
  float* ws  = (float*)d_ws;

  // Output layout (flat, return order): M_new, state, ll, rr
  const size_t NMD = (size_t)NB * DIN * DOUT;       // 16,777,216
  float* out_M     = out;
  float* out_state = out + NMD;
  float* out_ll    = out + NMD + 1;
  float* out_rr    = out_ll + (size_t)NB * DIN;

  // Workspace layout (~97 MB of floats)
  float* UU   = ws;                                 // [NB,DIN,DOUT] -> later A
  float* invL = ws + NMD;                           // [NB,DIN]
  float* invR = invL + (size_t)NB * DIN;            // [NB,DOUT]
  float* T    = invR + (size_t)NB * DOUT;           // [NB,DOUT,DOUT] -> later G
  float* Rinv = T + (size_t)NB * DOUT * DOUT;       // [NB,DOUT,DOUT]

  // 1) row stats: ll, ll^{-1/4}
  k_rowstats<<<NB * DIN, 256, 0, stream>>>(grad, L0, out_ll, invL);
  // 2) col stats: rr, rr^{-1/4} (+ state passthrough)
  k_colstats<<<NB, 256, 0, stream>>>(grad, R0, out_rr, invR, state, out_state);
  // 3) UU = grad scaled
  k_scale<<<NB * DIN, 256, 0, stream>>>(grad, invL, invR, UU);
  // 4) T = M^T UU          [DOUT x DOUT], K = DIN
  k_gemm_tn<<<NB * 8, 256, 0, stream>>>(M, UU, T);
  // 5) A = (M - LR2*UU) + M*(LR2*T)  -> overwrite UU in place
  k_gemm_nn<<<NB * 32, 256, 0, stream>>>(M, T, M, UU, -LR2, LR2, 1, UU);
  // 6) G = A^T A -> reuse T buffer
  k_gemm_tn<<<NB * 8, 256, 0, stream>>>(UU, UU, T);
  // 7) R = chol(G), Rinv = R^{-1}
  k_cholinv<<<NB, 256, 0, stream>>>(T, Rinv);
  // 8) M_new = Q = A * Rinv  (CholeskyQR; diag(R)>0 matches sign convention)
  k_gemm_nn<<<NB * 32, 256, 0, stream>>>(UU, Rinv, nullptr, nullptr, 0.f, 1.f,
                                         0, out_M);
}